// DeeperGCN_39006892982955
// MI455X (gfx1250) — compile-verified
//
#include <hip/hip_runtime.h>
#include <cstddef>

#define NN   10000
#define EE   160000
#define INC  128
#define HH   64
#define OUTC 112
#define LL   28
#define LN_EPS  1e-5f
#define MSG_EPS 1e-7f
#define SM_EPS  1e-16f

typedef __attribute__((ext_vector_type(2))) float v2f;
typedef __attribute__((ext_vector_type(8))) float v8f;

// ---------------- order-preserving float<->uint for atomic max ----------------
__device__ __forceinline__ unsigned enc_ord(float f) {
    unsigned u = __float_as_uint(f);
    return (u & 0x80000000u) ? ~u : (u | 0x80000000u);
}
// untouched slots stay 0u -> decode to 0.0f (reference: non-finite max -> 0)
__device__ __forceinline__ float dec_mx(unsigned u) {
    if (u == 0u) return 0.0f;
    unsigned b = (u & 0x80000000u) ? (u & 0x7fffffffu) : ~u;
    return __uint_as_float(b);
}

__device__ __forceinline__ void atomAddF(float* p, float v) {
    unsafeAtomicAdd(p, v);   // hardware global_atomic_add_f32
}

// ---------------- LayerNorm + ReLU, one wave (32 lanes) per row ----------------
// C must be 64 or 128 (per-lane 2 or 4 channels).
__global__ void ln_relu_kernel(const float* __restrict__ X,
                               const float* __restrict__ g,
                               const float* __restrict__ b,
                               float* __restrict__ Y, int n, int C) {
    int w    = (int)((blockIdx.x * blockDim.x + threadIdx.x) >> 5);
    int lane = threadIdx.x & 31;
    if (w >= n) return;
    const float* xr = X + (size_t)w * C;
    int per = C >> 5;                   // 2 or 4
    float vals[4];
    float s = 0.f;
    for (int i = 0; i < per; ++i) {
        float v = xr[lane + (i << 5)];
        vals[i] = v; s += v;
    }
    for (int off = 16; off; off >>= 1) s += __shfl_xor(s, off, 32);
    float mu = s / (float)C;
    float s2 = 0.f;
    for (int i = 0; i < per; ++i) { float d = vals[i] - mu; s2 += d * d; }
    for (int off = 16; off; off >>= 1) s2 += __shfl_xor(s2, off, 32);
    float rs = rsqrtf(s2 / (float)C + LN_EPS);
    for (int i = 0; i < per; ++i) {
        int c = lane + (i << 5);
        float y = (vals[i] - mu) * rs * g[c] + b[c];
        Y[(size_t)w * C + c] = fmaxf(y, 0.f);
    }
}

// ---------------- per-layer aggregation init: mxu=0, den=0, agg=z ----------------
// Seeding agg with z implements GENConv's out = aggr + x_r without a second
// GEMM A-operand (keeps the WMMA inner loop branch-free).
__global__ void agg_init_kernel(unsigned* __restrict__ mxu, float* __restrict__ den,
                                float* __restrict__ agg, const float* __restrict__ z,
                                int n) {
    int i = (int)(blockIdx.x * blockDim.x + threadIdx.x);
    if (i < n) { mxu[i] = 0u; den[i] = 0.f; agg[i] = z[i]; }
}

// ---------------- edge passes: one wave per edge, 2 channels per lane ----------------
__global__ void edge_max_kernel(const float* __restrict__ z, const int* __restrict__ ei,
                                const float* __restrict__ tptr, unsigned* __restrict__ mxu) {
    int w = (int)((blockIdx.x * blockDim.x + threadIdx.x) >> 5);
    if (w >= EE) return;
    int lane = threadIdx.x & 31;
    int src = ei[w], dst = ei[EE + w];
    float t = *tptr;
    int c = lane << 1;
    float2 zz = *(const float2*)(z + (size_t)src * HH + c);
    unsigned* mp = mxu + (size_t)dst * HH + c;
    atomicMax(mp + 0, enc_ord((zz.x + MSG_EPS) * t));
    atomicMax(mp + 1, enc_ord((zz.y + MSG_EPS) * t));
}

__global__ void edge_expsum_kernel(const float* __restrict__ z, const int* __restrict__ ei,
                                   const float* __restrict__ tptr,
                                   const unsigned* __restrict__ mxu, float* __restrict__ den) {
    int w = (int)((blockIdx.x * blockDim.x + threadIdx.x) >> 5);
    if (w >= EE) return;
    int lane = threadIdx.x & 31;
    int src = ei[w], dst = ei[EE + w];
    float t = *tptr;
    int c = lane << 1;
    float2 zz = *(const float2*)(z + (size_t)src * HH + c);
    size_t base = (size_t)dst * HH + c;
    float mx0 = dec_mx(mxu[base + 0]);
    float mx1 = dec_mx(mxu[base + 1]);
    atomAddF(den + base + 0, expf((zz.x + MSG_EPS) * t - mx0));
    atomAddF(den + base + 1, expf((zz.y + MSG_EPS) * t - mx1));
}

__global__ void edge_agg_kernel(const float* __restrict__ z, const int* __restrict__ ei,
                                const float* __restrict__ tptr,
                                const unsigned* __restrict__ mxu,
                                const float* __restrict__ den, float* __restrict__ agg) {
    int w = (int)((blockIdx.x * blockDim.x + threadIdx.x) >> 5);
    if (w >= EE) return;
    int lane = threadIdx.x & 31;
    int src = ei[w], dst = ei[EE + w];
    float t = *tptr;
    int c = lane << 1;
    float2 zz = *(const float2*)(z + (size_t)src * HH + c);
    size_t base = (size_t)dst * HH + c;
    float m0 = zz.x + MSG_EPS, m1 = zz.y + MSG_EPS;
    float e0 = expf(m0 * t - dec_mx(mxu[base + 0]));
    float e1 = expf(m1 * t - dec_mx(mxu[base + 1]));
    float a0 = e0 / (den[base + 0] + SM_EPS);
    float a1 = e1 / (den[base + 1] + SM_EPS);
    atomAddF(agg + base + 0, a0 * m0);
    atomAddF(agg + base + 1, a1 * m1);
}

// ---------------- WMMA f32 GEMM: D = A @ W + bias [+ Cres] ----------------
// K, N compile-time -> all strides are immediates, K-loop unrolls into load
// clauses + back-to-back v_wmma_f32_16x16x4_f32.
// Layout (V_WMMA_F32_16X16X4_F32): A frag: row = lane&15, K pair = 2*(lane>>4)+{0,1};
// B frag: col = lane&15, same K pair; C/D: row = v + 8*(lane>>4), col = lane&15.
template <int K, int N>
__global__ void gemm_wmma_f32(const float* __restrict__ A,
                              const float* __restrict__ W,
                              const float* __restrict__ bias,
                              const float* __restrict__ Cres, float* __restrict__ D,
                              int M) {
    int wv   = threadIdx.x >> 5;
    int lane = threadIdx.x & 31;
    int row0 = (int)blockIdx.x << 4;
    int n0   = (((int)blockIdx.y << 2) + wv) << 4;
    if (row0 >= M || n0 >= N) return;   // wave-uniform exit: EXEC all-1s at WMMA
    int half = lane >> 4;
    int lm   = lane & 15;
    int colv = n0 + lm;

    v8f acc;
    float bse = bias[colv];
    if (Cres) {
#pragma unroll
        for (int v = 0; v < 8; ++v)
            acc[v] = bse + Cres[(size_t)(row0 + v + (half << 3)) * N + colv];
    } else {
#pragma unroll
        for (int v = 0; v < 8; ++v) acc[v] = bse;
    }

    const float* ap = A + (size_t)(row0 + lm) * K + (half << 1);
    const float* wp = W + (size_t)(half << 1) * N + colv;
#pragma unroll 8
    for (int k0 = 0; k0 < K; k0 += 4) {
        v2f a, b;
        a.x = ap[0];
        a.y = ap[1];
        b.x = wp[0];
        b.y = wp[N];
        ap += 4;
        wp += 4 * N;
        acc = __builtin_amdgcn_wmma_f32_16x16x4_f32(false, a, false, b,
                                                    (short)0, acc, false, false);
    }

#pragma unroll
    for (int v = 0; v < 8; ++v)
        D[(size_t)(row0 + v + (half << 3)) * N + colv] = acc[v];
}

// ---------------- host-side orchestration ----------------
template <int K, int N>
static inline void launch_gemm(const float* A, const float* W, const float* bias,
                               const float* Cres, float* D, int M, hipStream_t s) {
    dim3 grid((M + 15) / 16, (N / 16 + 3) / 4);
    gemm_wmma_f32<K, N><<<grid, 128, 0, s>>>(A, W, bias, Cres, D, M);
}

extern "C" void kernel_launch(void* const* d_in, const int* in_sizes, int n_in,
                              void* d_out, int out_size, void* d_ws, size_t ws_size,
                              hipStream_t stream) {
    (void)in_sizes; (void)n_in; (void)out_size; (void)ws_size;
    const float* x      = (const float*)d_in[0];
    const int*   ei     = (const int*)d_in[1];
    const float* enc_W  = (const float*)d_in[2];
    const float* enc_b  = (const float*)d_in[3];
    const float* ln_g   = (const float*)d_in[4];
    const float* ln_b   = (const float*)d_in[5];
    const float* tvec   = (const float*)d_in[6];
    const float* W1     = (const float*)d_in[7];
    const float* b1     = (const float*)d_in[8];
    const float* mlp_g  = (const float*)d_in[9];
    const float* mlp_b  = (const float*)d_in[10];
    const float* W2     = (const float*)d_in[11];
    const float* b2     = (const float*)d_in[12];
    const float* head_W = (const float*)d_in[13];
    const float* head_b = (const float*)d_in[14];
    float* out = (float*)d_out;

    // workspace layout (floats): all regions fully rewritten each call
    float* ws  = (float*)d_ws;
    float* h   = ws;                        // [N,64]
    float* z   = h   + (size_t)NN * HH;     // [N,64]
    float* den = z   + (size_t)NN * HH;     // [N,64]
    float* agg = den + (size_t)NN * HH;     // [N,64]
    float* tmp = agg + (size_t)NN * HH;     // [N,128]
    float* u   = tmp + (size_t)NN * 2 * HH; // [N,128]
    unsigned* mxu = (unsigned*)(u + (size_t)NN * 2 * HH); // [N,64]

    const int lnBlocks   = (NN * 32 + 255) / 256;
    const int initBlocks = (NN * HH + 255) / 256;
    const int edgeBlocks = (EE * 32 + 255) / 256;

    // node encoder: h = x @ enc_W + enc_b
    launch_gemm<INC, HH>(x, enc_W, enc_b, nullptr, h, NN, stream);

    for (int l = 0; l < LL; ++l) {
        // z = relu(LN(h))
        ln_relu_kernel<<<lnBlocks, 256, 0, stream>>>(h, ln_g + (size_t)l * HH,
                                                     ln_b + (size_t)l * HH, z, NN, HH);
        // mxu=0 (ordered -inf), den=0, agg=z (residual of softmax aggr)
        agg_init_kernel<<<initBlocks, 256, 0, stream>>>(mxu, den, agg, z, NN * HH);
        // softmax aggregation over edges
        edge_max_kernel   <<<edgeBlocks, 256, 0, stream>>>(z, ei, tvec + l, mxu);
        edge_expsum_kernel<<<edgeBlocks, 256, 0, stream>>>(z, ei, tvec + l, mxu, den);
        edge_agg_kernel   <<<edgeBlocks, 256, 0, stream>>>(z, ei, tvec + l, mxu, den, agg);
        // tmp = agg @ W1 + b1          (agg already holds aggr + z)
        launch_gemm<HH, 2 * HH>(agg, W1 + (size_t)l * HH * 2 * HH,
                                b1 + (size_t)l * 2 * HH, nullptr, tmp, NN, stream);
        // u = relu(LN(tmp))
        ln_relu_kernel<<<lnBlocks, 256, 0, stream>>>(tmp, mlp_g + (size_t)l * 2 * HH,
                                                     mlp_b + (size_t)l * 2 * HH, u, NN, 2 * HH);
        // h = h + u @ W2 + b2   (residual folded into C-init)
        launch_gemm<2 * HH, HH>(u, W2 + (size_t)l * 2 * HH * HH,
                                b2 + (size_t)l * HH, h, h, NN, stream);
    }

    // head: out = h @ head_W + head_b
    launch_gemm<HH, OUTC>(h, head_W, head_b, nullptr, out, NN, stream);
}